// BoundaryLoss_10445360464195
// MI455X (gfx1250) — compile-verified
//
#include <hip/hip_runtime.h>
#include <math.h>

#define BB 8
#define CC 3
#define HH 256
#define WW 256
#define NTOT (BB*CC*HH*WW)    // 1572864
#define NIMG (BB*CC)          // 24
#define NROWS (BB*CC*HH)      // 6144
#define WTILES (WW/16)        // 16
#define RBLOCKS 1024

typedef __attribute__((ext_vector_type(2))) float v2f;
typedef __attribute__((ext_vector_type(8))) float v8f;

// ---------------- init: zero per-image any_pos flags ----------------
__global__ void bl_init_kernel(int* anyPos) {
    int t = threadIdx.x;
    if (t < NIMG) anyPos[t] = 0;
}

// ---------------- pass 1: per-row 1D clipped squared distances ----------------
// g2fg[r][j] = min(dist to nearest gt>0.5 in row, 1e4)^2  (for negdis)
// g2bg[r][j] = min(dist to nearest gt<=0.5 in row, 1e4)^2 (for posdis)
__global__ void bl_rowdist_kernel(const float* __restrict__ gt,
                                  float* __restrict__ g2fg,
                                  float* __restrict__ g2bg,
                                  int* __restrict__ anyPos) {
    int r = blockIdx.x * blockDim.x + threadIdx.x;
    if (r >= NROWS) return;
    int base = r * WW;
    int lastf = -(1 << 26), lastb = -(1 << 26);
    bool any = false;
    for (int j = 0; j < WW; ++j) {
        float gv = gt[base + j];
        bool pos = gv > 0.5f;
        if (pos) { lastf = j; any = true; } else { lastb = j; }
        g2fg[base + j] = (float)(j - lastf);   // huge if none yet; clipped later
        g2bg[base + j] = (float)(j - lastb);
    }
    int nextf = (1 << 26), nextb = (1 << 26);
    for (int j = WW - 1; j >= 0; --j) {
        float gv = gt[base + j];
        bool pos = gv > 0.5f;
        if (pos) nextf = j; else nextb = j;
        float df = fminf(fminf(g2fg[base + j], (float)(nextf - j)), 10000.0f);
        float db = fminf(fminf(g2bg[base + j], (float)(nextb - j)), 10000.0f);
        g2fg[base + j] = df * df;
        g2bg[base + j] = db * db;
    }
    if (any) atomicOr(&anyPos[r / HH], 1);   // idempotent -> deterministic
}

// ---------------- pass 2: min-plus convolution along H, sdf in place ----------------
// One block handles one (image, 16-wide w tile). Tiles are filled with CDNA5
// async copies (memory -> LDS directly, no VGPR staging), tracked by ASYNCcnt.
__global__ void bl_edt_kernel(float* __restrict__ g2fg,          // in: g2fg, out: sdf
                              const float* __restrict__ g2bg,
                              const int* __restrict__ anyPos) {
    __shared__ __align__(16) float s_fg[HH * 16];                 // 16 KB
    __shared__ __align__(16) float s_bg[HH * 16];                 // 16 KB
    int wt    = blockIdx.x & (WTILES - 1);
    int img   = blockIdx.x >> 4;                                  // WTILES == 16
    int cbase = img * (HH * WW);
    int w0    = wt * 16;
    int tid   = threadIdx.x;

    // Async fill: 1024 B128 chunks per plane (16 floats/row = 4 chunks), 256 threads.
    // Global addr is 16B aligned per chunk; LDS dst is 16B aligned.
    unsigned fgbase = (unsigned)(size_t)s_fg;
    unsigned bgbase = (unsigned)(size_t)s_bg;
    for (int t = tid; t < (HH * 16) / 4; t += 256) {
        int j   = t >> 2;
        int c4  = (t & 3) << 2;
        int goff = cbase + (j << 8) + w0 + c4;
        unsigned long long ga_fg = (unsigned long long)(size_t)(g2fg + goff);
        unsigned long long ga_bg = (unsigned long long)(size_t)(g2bg + goff);
        unsigned l_fg = fgbase + ((unsigned)t << 4);
        unsigned l_bg = bgbase + ((unsigned)t << 4);
        asm volatile("global_load_async_to_lds_b128 %0, %1, off"
                     :: "v"(l_fg), "v"(ga_fg) : "memory");
        asm volatile("global_load_async_to_lds_b128 %0, %1, off"
                     :: "v"(l_bg), "v"(ga_bg) : "memory");
    }
    asm volatile("s_wait_asynccnt 0x0" ::: "memory");             // own copies done
    __syncthreads();                                              // all waves' copies done

    int ap = anyPos[img];
    int wl = tid & 15;
    int q  = tid >> 4;                                            // 0..15
    for (int k = 0; k < 16; ++k) {
        int i = q * 16 + k;
        float fi = (float)i;
        float m1 = 3.0e38f, m2 = 3.0e38f;
#pragma unroll 4
        for (int j = 0; j < HH; ++j) {
            float d  = fi - (float)j;
            float t2 = d * d;
            int li = (j << 4) + wl;
            m1 = fminf(m1, s_fg[li] + t2);                        // toward negdis
            m2 = fminf(m2, s_bg[li] + t2);                        // toward posdis
        }
        float sdfv = ap ? (sqrtf(m1) - sqrtf(m2)) : 0.0f;
        g2fg[cbase + (i << 8) + w0 + wl] = sdfv;                  // in-place: sdf
    }
}

// ---------------- coarse mask: 7x7 count-normalized window sum ----------------
__global__ void bl_coarse_kernel(const float* __restrict__ gt,
                                 float* __restrict__ coarse) {
    int idx = blockIdx.x * blockDim.x + threadIdx.x;
    if (idx >= NTOT) return;
    int w = idx & (WW - 1);
    int h = (idx >> 8) & (HH - 1);
    int imgbase = idx & ~(HH * WW - 1);
    float gv = gt[idx];
    int hlo = max(h - 3, 0), hhi = min(h + 3, HH - 1);
    int wlo = max(w - 3, 0), whi = min(w + 3, WW - 1);
    float s = 0.0f;
    for (int hh = hlo; hh <= hhi; ++hh) {
        int rb = imgbase + (hh << 8);
        for (int ww = wlo; ww <= whi; ++ww) s += gt[rb + ww];
    }
    float cnt = (float)((hhi - hlo + 1) * (whi - wlo + 1));
    float nbr = s / cnt * 49.0f;
    coarse[idx] = (nbr > 25.0f && gv > 0.5f) ? 1.0f : 0.0f;
}

// ---------------- weighted mean reduce: dilate + sigmoid + WMMA wave sum ----------------
__global__ void bl_reduce_kernel(const float* __restrict__ pred,
                                 const float* __restrict__ sdf,
                                 const float* __restrict__ coarse,
                                 float* __restrict__ partials) {
    __shared__ float wsum[8];
    int tid = threadIdx.x;
    int gthreads = gridDim.x * blockDim.x;
    int idx = blockIdx.x * blockDim.x + tid;
    const float invn = 1.0f / (256.0f + 1e-6f);
    float accx = 0.0f, accy = 0.0f;
    int iters = (NTOT + gthreads - 1) / gthreads;                 // uniform trip count
    for (int it = 0; it < iters; ++it, idx += gthreads) {
        if (idx + gthreads < NTOT) {                              // cover next iteration
            __builtin_prefetch(&pred[idx + gthreads], 0, 0);      // global_prefetch_b8
            __builtin_prefetch(&sdf[idx + gthreads], 0, 0);
        }
        float val = 0.0f;
        if (idx < NTOT) {
            int w = idx & (WW - 1);
            int h = (idx >> 8) & (HH - 1);
            int imgbase = idx & ~(HH * WW - 1);
            int hlo = max(h - 3, 0), hhi = min(h + 3, HH - 1);
            int wlo = max(w - 3, 0), whi = min(w + 3, WW - 1);
            float m = 0.0f;                                       // 7x7 max dilation
            for (int hh = hlo; hh <= hhi; ++hh) {
                int rb = imgbase + (hh << 8);
                for (int ww = wlo; ww <= whi; ++ww) m = fmaxf(m, coarse[rb + ww]);
            }
            float x = pred[idx];
            float p = 1.0f / (1.0f + expf(-x));
            val = sdf[idx] * invn * p * m;
        }
        if (it & 1) accy += val; else accx += val;
    }
    // Wave reduction via V_WMMA_F32_16X16X4_F32: D = A(16x4) * ones(4x16).
    // All-ones B makes the result layout-independent: every D column holds the
    // 16 row-sums of A; summing a lane's 8 C regs + shfl_xor(16) = wave total.
    v2f a;   a.x = accx; a.y = accy;
    v2f one; one.x = 1.0f; one.y = 1.0f;
    v8f c = {};
    c = __builtin_amdgcn_wmma_f32_16x16x4_f32(false, a, false, one,
                                              (short)0, c, false, false);
    float s = c[0] + c[1] + c[2] + c[3] + c[4] + c[5] + c[6] + c[7];
    s += __shfl_xor(s, 16, 32);
    if ((tid & 31) == 0) wsum[tid >> 5] = s;
    __syncthreads();
    if (tid == 0) {
        float t = 0.0f;
        for (int i = 0; i < 8; ++i) t += wsum[i];
        partials[blockIdx.x] = t;
    }
}

// ---------------- final: deterministic partial sum -> mean ----------------
__global__ void bl_final_kernel(const float* __restrict__ partials,
                                float* __restrict__ out) {
    __shared__ float l[256];
    int tid = threadIdx.x;
    float s = 0.0f;
    for (int i = tid; i < RBLOCKS; i += 256) s += partials[i];
    l[tid] = s;
    __syncthreads();
    if (tid == 0) {
        float t = 0.0f;
        for (int i = 0; i < 256; ++i) t += l[i];
        out[0] = t / (float)NTOT;
    }
}

extern "C" void kernel_launch(void* const* d_in, const int* in_sizes, int n_in,
                              void* d_out, int out_size, void* d_ws, size_t ws_size,
                              hipStream_t stream) {
    const float* pred = (const float*)d_in[0];
    const float* gt   = (const float*)d_in[1];
    float* out = (float*)d_out;

    float* g2fg     = (float*)d_ws;            // N floats; becomes sdf in place
    float* g2bg     = g2fg + NTOT;             // N floats
    float* coarse   = g2bg + NTOT;             // N floats
    float* partials = coarse + NTOT;           // RBLOCKS floats
    int*   anyPos   = (int*)(partials + RBLOCKS); // NIMG ints

    bl_init_kernel<<<1, 32, 0, stream>>>(anyPos);
    bl_rowdist_kernel<<<(NROWS + 255) / 256, 256, 0, stream>>>(gt, g2fg, g2bg, anyPos);
    bl_edt_kernel<<<NIMG * WTILES, 256, 0, stream>>>(g2fg, g2bg, anyPos);
    bl_coarse_kernel<<<(NTOT + 255) / 256, 256, 0, stream>>>(gt, coarse);
    bl_reduce_kernel<<<RBLOCKS, 256, 0, stream>>>(pred, g2fg, coarse, partials);
    bl_final_kernel<<<1, 256, 0, stream>>>(partials, out);
}